// AELoss_78391743087244
// MI455X (gfx1250) — compile-verified
//
#include <hip/hip_runtime.h>

typedef __attribute__((ext_vector_type(2))) float v2f;
typedef __attribute__((ext_vector_type(8))) float v8f;

#define BATCH     8
#define N_PTS     4096
#define ROWS_PER_BLOCK 128
#define THREADS   256
#define CHUNKS    (N_PTS / ROWS_PER_BLOCK)   // 32 row-chunks per batch
#define BLOCKS_PER_DIR (BATCH * CHUNKS)      // 256

__global__ void chamfer_zero_kernel(float* out, int n) {
    int i = blockIdx.x * blockDim.x + threadIdx.x;
    if (i < n) out[i] = 0.0f;
}

// Chamfer distance, both directions in one launch.
// Distance tile via WMMA f32 16x16x4:
//   A[row] = (-2px, -2py, -2pz, 1)          (16x4, stationary per wave)
//   B[col] = ( qx,   qy,   qz,  ||q||^2)    (4x16, pre-swizzled in LDS)
//   C[row] = ||p||^2                        (constant across the j sweep)
//   D = A x B + C  ==  ||p||^2 + ||q||^2 - 2 p.q
//
// LDS holds B already in the ISA's 4x16 f32 operand layout: for tile jt,
// lane l<16 reads (K0,K1) = (x,y) of column l, lane l>=16 reads
// (K2,K3) = (z, ||q||^2) of column l-16. Inner loop per 256 distances:
// 1x ds_load_b64 + 1x v_wmma + 4x v_min3 (no per-iteration selects).
__global__ __launch_bounds__(THREADS)
void chamfer_kernel(const float* __restrict__ ref,    // [B,N,3]
                    const float* __restrict__ rec,    // [B,N,3]
                    float* __restrict__ out, int out_size)
{
    __shared__ float2 qs[N_PTS * 2];   // 64 KB, B-operand swizzled tiles

    const int dir   = blockIdx.x / BLOCKS_PER_DIR;   // 0: ref->rec, 1: rec->ref
    const int blk   = blockIdx.x % BLOCKS_PER_DIR;
    const int b     = blk / CHUNKS;
    const int chunk = blk % CHUNKS;
    const float* P  = dir ? rec : ref;   // row points
    const float* Q  = dir ? ref : rec;   // column points

    const int tid  = (int)threadIdx.x;
    const int wave = tid >> 5;
    const int lane = tid & 31;
    const int half = lane >> 4;          // 0: lanes 0-15, 1: lanes 16-31
    const int l15  = lane & 15;

    // ---- stage column cloud into LDS, pre-swizzled to WMMA B layout ----
    const float* Qb = Q + (size_t)b * N_PTS * 3;
    for (int idx = tid; idx < N_PTS; idx += THREADS) {
        float qx = Qb[idx * 3 + 0];
        float qy = Qb[idx * 3 + 1];
        float qz = Qb[idx * 3 + 2];
        int jt = idx >> 4, c = idx & 15;
        qs[jt * 32 + c]      = make_float2(qx, qy);                    // K0,K1
        qs[jt * 32 + 16 + c] = make_float2(qz, qx*qx + qy*qy + qz*qz); // K2,K3
    }
    __syncthreads();

    // ---- stationary A matrix (16x4 f32 layout: V0={K0|K2}, V1={K1|K3}) ----
    const int rowBase = chunk * ROWS_PER_BLOCK + wave * 16;
    const float* Pp = P + ((size_t)b * N_PTS + (rowBase + l15)) * 3;
    float px = Pp[0], py = Pp[1], pz = Pp[2];
    float pn = px * px + py * py + pz * pz;   // norm of row (rowBase + l15)

    v2f a;
    a.x = half ? (-2.0f * pz) : (-2.0f * px);   // K2 : K0
    a.y = half ? 1.0f         : (-2.0f * py);   // K3 : K1  (K3 multiplies ||q||^2)

    // C matrix: row norms. C/D layout: VGPR k <-> row k (lanes 0-15) or
    // row k+8 (lanes 16-31). Gather with lane shuffles.
    v8f cmat;
#pragma unroll
    for (int k = 0; k < 8; ++k)
        cmat[k] = __shfl(pn, (half << 3) + k, 32);

    // ---- sweep all 256 column tiles, keep running elementwise min ----
    v8f m;
#pragma unroll
    for (int k = 0; k < 8; ++k) m[k] = 3.4e38f;

    for (int jt = 0; jt < N_PTS / 16; ++jt) {
        float2 q = qs[jt * 32 + lane];   // already in B operand layout
        v2f bm;
        bm.x = q.x;
        bm.y = q.y;
        v8f d = __builtin_amdgcn_wmma_f32_16x16x4_f32(
            false, a, false, bm, (short)0, cmat, false, false);
#pragma unroll
        for (int k = 0; k < 8; ++k) m[k] = fminf(m[k], d[k]);
    }

    // ---- per-row minima: min across the 16 lanes of each half ----
#pragma unroll
    for (int k = 0; k < 8; ++k) {
        float v = m[k];
        v = fminf(v, __shfl_xor(v, 1, 32));
        v = fminf(v, __shfl_xor(v, 2, 32));
        v = fminf(v, __shfl_xor(v, 4, 32));
        v = fminf(v, __shfl_xor(v, 8, 32));
        m[k] = v;
    }
    float s = 0.0f;
#pragma unroll
    for (int k = 0; k < 8; ++k) s += m[k];
    s += __shfl_xor(s, 16, 32);     // rows 0-7 (half 0) + rows 8-15 (half 1)

    if (lane == 0) {
        atomicAdd(&out[b], s);                        // "Criterion"
        if (out_size >= 2 * BATCH)
            atomicAdd(&out[BATCH + b], s);            // "Chamfer" duplicate
    }
}

extern "C" void kernel_launch(void* const* d_in, const int* in_sizes, int n_in,
                              void* d_out, int out_size, void* d_ws, size_t ws_size,
                              hipStream_t stream) {
    const float* ref_cloud = (const float*)d_in[0];
    const float* recon     = (const float*)d_in[1];
    float* out = (float*)d_out;

    chamfer_zero_kernel<<<(out_size + 63) / 64, 64, 0, stream>>>(out, out_size);

    dim3 grid(2 * BLOCKS_PER_DIR);   // both directions in one launch
    dim3 block(THREADS);
    chamfer_kernel<<<grid, block, 0, stream>>>(ref_cloud, recon, out, out_size);
}